// SpDepthWSepaConv3d_43533788512900
// MI455X (gfx1250) — compile-verified
//
#include <hip/hip_runtime.h>
#include <stdint.h>

typedef unsigned int v4u __attribute__((ext_vector_type(4)));
typedef int          v4i __attribute__((ext_vector_type(4)));
typedef int          v8i __attribute__((ext_vector_type(8)));

#define KVOL 27
#define CH   64
#define NW   (KVOL * CH)      /* 1728 weight floats = 6912 B */
#define VOXB 64               /* voxels per block */
#define THREADS 256

__global__ __launch_bounds__(THREADS)
void SpDepthWSepaConv3d_kernel(const float* __restrict__ feat,
                               const float* __restrict__ weight,
                               const float* __restrict__ bias,
                               const int*   __restrict__ pin,
                               const unsigned char* __restrict__ pmask,
                               float* __restrict__ out,
                               int n)
{
    __shared__ __align__(16) float sW[NW];

    // ---- Stage the 27x64 weight table into LDS with one TDM DMA op ----
    // (wave 0, lane 0 issues; TDM ignores EXEC; other waves skip via execz)
    if (threadIdx.x == 0) {
        uint64_t ga  = (uint64_t)(uintptr_t)weight;
        uint32_t lds = (uint32_t)(uintptr_t)&sW[0];

        v4u g0;
        g0[0] = 1u;                                   // count=1 (valid), user mode
        g0[1] = lds;                                  // lds_addr (bytes)
        g0[2] = (uint32_t)ga;                         // global_addr[31:0]
        g0[3] = ((uint32_t)(ga >> 32) & 0x1FFFFFFu)   // global_addr[56:32]
              | (2u << 30);                           // type = 2 ("image")

        v8i g1;
        g1[0] = 0x00020000;                 // data_size = 2 (4-byte elems)
        g1[1] = (int)((NW & 0xFFFF) << 16); // tensor_dim0[15:0] = 1728
        g1[2] = (int)(1u << 16);            // tensor_dim0[31:16]=0 | tensor_dim1=1
        g1[3] = (int)((NW & 0xFFFF) << 16); // tile_dim0 = 1728
        g1[4] = 1;                          // tile_dim1 = 1, tile_dim2 = 0
        g1[5] = NW;                         // tensor_dim0_stride[31:0] = 1728
        g1[6] = 0;
        g1[7] = 0;

        v4i g2 = {0, 0, 0, 0};              // D# group 2 (unused dims)
        v4i g3 = {0, 0, 0, 0};              // D# group 3 (unused dims)
        v8i g4 = {0, 0, 0, 0, 0, 0, 0, 0};  // extra group for 6-arg builtin

        __builtin_amdgcn_tensor_load_to_lds(g0, g1, g2, g3, g4, 0);
        __builtin_amdgcn_s_wait_tensorcnt(0);
    }
    __syncthreads();

    const float4* __restrict__ sW4 = (const float4*)sW;
    const float4* __restrict__ f4  = (const float4*)feat;
    const float4* __restrict__ b4  = (const float4*)bias;
    float4*       __restrict__ o4  = (float4*)out;

    const int lvox = threadIdx.x >> 4;   // 0..15: which voxel in this pass
    const int cq   = threadIdx.x & 15;   // channel quad: 4 channels per lane
    const int vb   = (int)blockIdx.x * VOXB;

    #pragma unroll
    for (int it = 0; it < VOXB / 16; ++it) {
        const int v = vb + it * 16 + lvox;
        if (v < n) {
            float ax = 0.f, ay = 0.f, az = 0.f, aw = 0.f;

            for (int k = 0; k < KVOL; ++k) {
                // ~97% of neighbor taps are absent: test mask first so the
                // index load and the 16B feature gather are both skipped.
                if (pmask[k * n + v] != 0) {
                    const int idx = pin[k * n + v];
                    // 16 lanes x float4 = one contiguous 256B feature row
                    float4 f = f4[(size_t)idx * (CH / 4) + cq];
                    float4 w = sW4[k * (CH / 4) + cq];   // ds_load_b128 broadcast
                    ax = fmaf(f.x, w.x, ax);
                    ay = fmaf(f.y, w.y, ay);
                    az = fmaf(f.z, w.z, az);
                    aw = fmaf(f.w, w.w, aw);
                }
            }

            const float4 b = b4[cq];
            float4 r;
            r.x = ax + b.x;
            r.y = ay + b.y;
            r.z = az + b.z;
            r.w = aw + b.w;
            o4[(size_t)v * (CH / 4) + cq] = r;   // coalesced b128 store
        }
    }
}

extern "C" void kernel_launch(void* const* d_in, const int* in_sizes, int n_in,
                              void* d_out, int out_size, void* d_ws, size_t ws_size,
                              hipStream_t stream)
{
    (void)n_in; (void)out_size; (void)d_ws; (void)ws_size;

    const float*         feat   = (const float*)d_in[0];          // [N, 64]
    const float*         weight = (const float*)d_in[1];          // [27, 64, 1, 1]
    const float*         bias   = (const float*)d_in[2];          // [64]
    const int*           pin    = (const int*)d_in[3];            // [27, N] int32
    /* d_in[4] = pair_out: identity where mask is set -> gather form, unused */
    const unsigned char* pmask  = (const unsigned char*)d_in[5];  // [27, N] bool

    const int N = in_sizes[0] / CH;   // 150000

    dim3 grid((N + VOXB - 1) / VOXB);
    SpDepthWSepaConv3d_kernel<<<grid, THREADS, 0, stream>>>(
        feat, weight, bias, pin, pmask, (float*)d_out, N);
}